// SphConv2d_24103356465399
// MI455X (gfx1250) — compile-verified
//
#include <hip/hip_runtime.h>

// Problem constants (from the reference): B=4, C=64, OUT=64, N=163842, K2=9
#define B_    4
#define C_    64
#define OUT_  64
#define N_    163842
#define K2_   9
#define KTOT  (C_ * K2_)            // 576  (K order: k = k2*64 + c)
#define WROW  580                   // row stride (dwords): 576 + 4, == 4 mod 64 banks
#define NT    32                    // output columns per workgroup tile
#define NTILES ((N_ + NT - 1) / NT) // 5121
#define TOTAL_TILES (B_ * NTILES)   // 20484
#define MAIN_GRID 2048
#define SHMEM_BYTES ((OUT_ + NT) * WROW * 4) // 222720 <= 320KB

typedef float v2f __attribute__((ext_vector_type(2)));
typedef float v8f __attribute__((ext_vector_type(8)));

#if defined(__has_builtin)
#if __has_builtin(__builtin_amdgcn_global_load_async_to_lds_b128) && \
    __has_builtin(__builtin_amdgcn_s_wait_asynccnt)
#define USE_ASYNC_LDS 1
#endif
#endif

#ifdef USE_ASYNC_LDS
// Parameter types per the compiler diagnostic: vector-of-4-int pointers in
// global (addrspace 1) and LDS (addrspace 3) address spaces.
typedef int v4i_vs __attribute__((__vector_size__(16)));
typedef __attribute__((address_space(1))) v4i_vs* async_gptr;
typedef __attribute__((address_space(3))) v4i_vs* async_lptr;
#endif

// ---------------------------------------------------------------------------
// Pass 1: transpose fmap (B, C, N) -> fmapT (B, N+1, C), channel-last, so a
// gathered neighbor is one contiguous 256B row. Row N is the zero pad row.
// ---------------------------------------------------------------------------
__global__ __launch_bounds__(256) void transpose_pad(const float* __restrict__ fmap,
                                                     float* __restrict__ fmapT) {
  __shared__ float tile[64][65];
  const int nchunks = (N_ + 63) / 64;
  const int b  = blockIdx.x / nchunks;
  const int n0 = (blockIdx.x % nchunks) * 64;
  const int t  = threadIdx.x;

  {
    const int nn = t & 63;
    const int cq = t >> 6;
    for (int c0 = 0; c0 < 64; c0 += 4) {
      const int c = c0 + cq;
      const int n = n0 + nn;
      tile[c][nn] = (n < N_) ? fmap[((size_t)b * C_ + c) * (size_t)N_ + n] : 0.0f;
    }
  }
  __syncthreads();
  {
    const int c  = t & 63;
    const int nq = t >> 6;
    for (int m0 = 0; m0 < 64; m0 += 4) {
      const int n = n0 + m0 + nq;
      if (n < N_)
        fmapT[((size_t)b * (N_ + 1) + n) * C_ + c] = tile[c][m0 + nq];
    }
  }
}

__global__ void zero_pad_row(float* __restrict__ fmapT) {
  const int t = blockIdx.x * blockDim.x + threadIdx.x;
  if (t < B_ * C_) {
    const int b = t >> 6, c = t & 63;
    fmapT[((size_t)b * (N_ + 1) + N_) * C_ + c] = 0.0f;
  }
}

// ---------------------------------------------------------------------------
// Pass 2: gathered GEMM with V_WMMA_F32_16X16X4_F32.
// Persistent workgroups: weight staged to LDS once (row-major A[o][k]),
// then loop over 32-column tiles; per tile stage a gathered B-tile
// (row-major B[col][k]) via async global->LDS copies, then 144 WMMAs/wave.
// ---------------------------------------------------------------------------
__global__ __launch_bounds__(256) void spconv_wmma(const float* __restrict__ fmapT,
                                                   const float* __restrict__ weight,
                                                   const float* __restrict__ bias,
                                                   const long long* __restrict__ nmap,
                                                   float* __restrict__ out) {
  extern __shared__ float lds[];
  float* __restrict__ Wlds = lds;               // [OUT_][WROW]  A row-major
  float* __restrict__ Bt   = lds + OUT_ * WROW; // [NT][WROW]    B row-major

  const int t = threadIdx.x;

  // Stage weight once. Coalesced global reads; LDS scatter to row-major
  // with K ordering k = k2*64 + c  (A[o][k] = weight[o][c][k2]).
  for (int idx = t; idx < OUT_ * KTOT; idx += 256) {
    const int o  = idx / KTOT;
    const int ck = idx - o * KTOT;
    const int c  = ck / K2_;
    const int k2 = ck - c * K2_;
    Wlds[o * WROW + k2 * C_ + c] = weight[idx];
  }

  const int lane  = t & 31;
  const int wave  = t >> 5;           // 8 waves
  const int mtile = (wave & 3) * 16;  // M tile: rows of OUT
  const int ctile = (wave >> 2) * 16; // column sub-tile within the 32
  const int half  = lane >> 4;        // K-interleave half (lanes 16..31)
  const int lm    = lane & 15;

  for (int tileId = blockIdx.x; tileId < TOTAL_TILES; tileId += gridDim.x) {
    const int b  = tileId / NTILES;
    const int n0 = (tileId - b * NTILES) * NT;

    __syncthreads(); // weight staged / previous compute done reading Bt

    // Stage B tile: 32 cols x 9 neighbors = 288 rows of 64 contiguous floats,
    // row r -> Bt[j][k2*64 .. k2*64+63] (contiguous in LDS too).
#ifdef USE_ASYNC_LDS
    {
      const int rloc  = t >> 4;       // 16 rows in flight, 16 lanes per row
      const int chunk = (t & 15) * 4; // 16B per lane
      for (int pass = 0; pass < 18; ++pass) {
        const int r  = pass * 16 + rloc;   // 0..287
        const int j  = r / K2_;
        const int k2 = r - j * K2_;
        const int n  = n0 + j;
        const long long idx = (n < N_) ? nmap[(size_t)n * K2_ + k2] : (long long)N_;
        const float* src = fmapT + ((size_t)b * (N_ + 1) + (size_t)idx) * C_ + chunk;
        float* dst = Bt + j * WROW + k2 * C_ + chunk;
        __builtin_amdgcn_global_load_async_to_lds_b128(
            (async_gptr)(void*)src, (async_lptr)(void*)dst, 0, 0);
      }
      __builtin_amdgcn_s_wait_asynccnt(0);
    }
#else
    {
      const int rloc = t >> 3;        // 32 rows in flight, 8 lanes per row
      const int cseg = (t & 7) * 8;   // 32B per lane
      for (int pass = 0; pass < 9; ++pass) {
        const int r  = pass * 32 + rloc;   // 0..287
        const int j  = r / K2_;
        const int k2 = r - j * K2_;
        const int n  = n0 + j;
        const long long idx = (n < N_) ? nmap[(size_t)n * K2_ + k2] : (long long)N_;
        const float* src = fmapT + ((size_t)b * (N_ + 1) + (size_t)idx) * C_ + cseg;
        const float4 v0 = *(const float4*)(src);
        const float4 v1 = *(const float4*)(src + 4);
        float* dst = Bt + j * WROW + k2 * C_ + cseg;
        *(float4*)(dst)     = v0;
        *(float4*)(dst + 4) = v1;
      }
    }
#endif
    __syncthreads();

    // 16x16 FP32 accumulator per wave; K = 576 in steps of 4.
    // Fragment loads are single ds_load_b64 per operand, bank-conflict-free.
    v8f acc = {0.f, 0.f, 0.f, 0.f, 0.f, 0.f, 0.f, 0.f};
    const float* Wr = Wlds + (mtile + lm) * WROW + 2 * half;
    const float* Br = Bt   + (ctile + lm) * WROW + 2 * half;
#pragma unroll 4
    for (int kk = 0; kk < KTOT / 4; ++kk) {
      const int kb = kk * 4;
      const v2f a  = *(const v2f*)(Wr + kb); // A[m][kb+2h], A[m][kb+2h+1]
      const v2f bb = *(const v2f*)(Br + kb); // B[kb+2h][n], B[kb+2h+1][n]
      acc = __builtin_amdgcn_wmma_f32_16x16x4_f32(
          /*neg_a=*/false, a, /*neg_b=*/false, bb,
          /*c_mod=*/(short)0, acc, /*reuse_a=*/false, /*reuse_b=*/false);
    }

    // Epilogue: bias + store. D layout: VGPR r -> M = mtile + r + 8*half,
    // N-col = ctile + (lane & 15).
    const int n = n0 + ctile + lm;
    if (n < N_) {
#pragma unroll
      for (int r = 0; r < 8; ++r) {
        const int m = mtile + r + half * 8;
        out[((size_t)b * OUT_ + m) * (size_t)N_ + n] = acc[r] + bias[m];
      }
    }
  }
}

// ---------------------------------------------------------------------------
extern "C" void kernel_launch(void* const* d_in, const int* in_sizes, int n_in,
                              void* d_out, int out_size, void* d_ws, size_t ws_size,
                              hipStream_t stream) {
  const float*      fmap   = (const float*)d_in[0];      // (B, C, 1, N) f32
  const float*      weight = (const float*)d_in[1];      // (OUT, C, K2) f32
  const float*      bias   = (const float*)d_in[2];      // (OUT,) f32
  const long long*  nmap   = (const long long*)d_in[3];  // (N*K2,) i64 in [0, N]
  float*            out    = (float*)d_out;              // (B, OUT, 1, N) f32
  float*            fmapT  = (float*)d_ws;               // (B, N+1, C) f32, ~160 MB

  const int nchunks = (N_ + 63) / 64;
  transpose_pad<<<dim3(B_ * nchunks), dim3(256), 0, stream>>>(fmap, fmapT);
  zero_pad_row<<<dim3(1), dim3(256), 0, stream>>>(fmapT);

  static_assert(SHMEM_BYTES <= 320 * 1024, "LDS over WGP budget");
  (void)hipFuncSetAttribute((const void*)spconv_wmma,
                            hipFuncAttributeMaxDynamicSharedMemorySize, SHMEM_BYTES);

  const int grid = (TOTAL_TILES < MAIN_GRID) ? TOTAL_TILES : MAIN_GRID;
  spconv_wmma<<<dim3(grid), dim3(256), SHMEM_BYTES, stream>>>(fmapT, weight, bias, nmap, out);
}